// SlotAttentionGMM_26053271617986
// MI455X (gfx1250) — compile-verified
//
#include <hip/hip_runtime.h>

typedef float v2f __attribute__((ext_vector_type(2)));
typedef float v8f __attribute__((ext_vector_type(8)));

constexpr int   cB  = 8;
constexpr int   cN  = 4096;
constexpr int   cD  = 128;
constexpr int   cNS = 8;
constexpr float cSCALE = 0.08838834764831845f;   // 128^-0.5
constexpr float cEPS   = 1e-8f;

__device__ __forceinline__ v8f zero8() {
  v8f z = {0.f,0.f,0.f,0.f,0.f,0.f,0.f,0.f};
  return z;
}

__device__ __forceinline__ v8f wmma4(v2f a, v2f b, v8f c) {
  // D = A(16x4) * B(4x16) + C, full fp32 (V_WMMA_F32_16X16X4_F32)
  return __builtin_amdgcn_wmma_f32_16x16x4_f32(false, a, false, b, (short)0, c,
                                               false, false);
}

// D(16x16) = A(16xK) @ W(16 rows x K)^T.  A row stride lda, W row stride ldw.
// Fragment layout per CDNA5 ISA 7.12.2: element i of lane l -> K = kk+i+2*(l>=16),
// row/col = l%16.
__device__ __forceinline__ v8f gemm_xWt(const float* A, int lda,
                                        const float* W, int ldw, int K) {
  const int l  = threadIdx.x & 31;
  const int hl = l >> 4;
  const int sub = l & 15;
  const float* ar = A + sub * lda + 2 * hl;
  const float* wr = W + sub * ldw + 2 * hl;
  v8f c = zero8();
  for (int kk = 0; kk < K; kk += 4) {
    v2f a = *(const v2f*)(ar + kk);
    v2f b = *(const v2f*)(wr + kk);
    c = wmma4(a, b, c);
  }
  return c;
}

// ---------------------------------------------------------------- init
__global__ void k_init(const float* __restrict__ slots_mu,
                       const float* __restrict__ slots_ls,
                       const float* __restrict__ noise,
                       float* __restrict__ slots, float* __restrict__ piw) {
  int t = blockIdx.x * blockDim.x + threadIdx.x;
  if (t < cB * cNS * 2 * cD) {
    int c = t & (2 * cD - 1);
    slots[t] = slots_mu[c] + expf(slots_ls[c]) * noise[t];
  }
  if (t < cB * 16) piw[t] = ((t & 15) < cNS) ? (1.0f / (float)cNS) : 0.f;
}

// ------------------------------------------- LN(inputs) -> k,v projections
__global__ void k_ln_proj(const float* __restrict__ inp, const float* __restrict__ g,
                          const float* __restrict__ bb, const float* __restrict__ Wk,
                          const float* __restrict__ Wv, float* __restrict__ kout,
                          float* __restrict__ vout) {
  __shared__ float xs[16 * 132];
  const int t = threadIdx.x;
  const int wv = t >> 5, l = t & 31, hl = l >> 4, sub = l & 15;
  const size_t base = (size_t)blockIdx.x * 16 * cD;
  for (int idx = t; idx < 16 * cD; idx += 256) {
    int r = idx >> 7, c = idx & 127;
    xs[r * 132 + c] = inp[base + idx];
  }
  __syncthreads();
  {
    // one row per 16-lane half-wave
    int row = wv * 2 + hl;
    float s = 0.f;
    for (int c = sub; c < cD; c += 16) s += xs[row * 132 + c];
    for (int o = 1; o < 16; o <<= 1) s += __shfl_xor(s, o, 32);
    float m = s * (1.f / cD);
    float vs = 0.f;
    for (int c = sub; c < cD; c += 16) { float d = xs[row * 132 + c] - m; vs += d * d; }
    for (int o = 1; o < 16; o <<= 1) vs += __shfl_xor(vs, o, 32);
    float inv = 1.f / sqrtf(vs * (1.f / cD) + 1e-5f);
    for (int c = sub; c < cD; c += 16)
      xs[row * 132 + c] = (xs[row * 132 + c] - m) * inv * g[c] + bb[c];
  }
  __syncthreads();
  const int colb = wv * 16;
  const float* ar  = xs + sub * 132 + 2 * hl;
  const float* wkr = Wk + (size_t)(colb + sub) * cD + 2 * hl;
  const float* wvr = Wv + (size_t)(colb + sub) * cD + 2 * hl;
  v8f ck = zero8(), cv = zero8();
  for (int kk = 0; kk < cD; kk += 4) {
    v2f a  = *(const v2f*)(ar + kk);
    v2f b1 = *(const v2f*)(wkr + kk);
    v2f b2 = *(const v2f*)(wvr + kk);
    ck = wmma4(a, b1, ck);
    cv = wmma4(a, b2, cv);
  }
  float* kd = kout + base + colb;
  float* vd = vout + base + colb;
  for (int r = 0; r < 8; r++) {
    int m = r + hl * 8;
    kd[m * cD + sub] = ck[r];
    vd[m * cD + sub] = cv[r];
  }
}

// ---------------------------------------- per-step: LN(slots), zero scratch
__global__ void k_slot_ln(const float* __restrict__ slots, const float* __restrict__ g,
                          const float* __restrict__ bb, float* __restrict__ slnw,
                          float* __restrict__ smuc, float* __restrict__ T1,
                          float* __restrict__ T2, float* __restrict__ rsum) {
  const int t = threadIdx.x, wv = t >> 5, l = t & 31;
  for (int row = wv; row < 64; row += 8) {
    const float* sr = slots + row * 256;
    float s = 0.f;
    for (int c = l; c < 256; c += 32) s += sr[c];
    for (int o = 1; o < 32; o <<= 1) s += __shfl_xor(s, o, 32);
    float m = s * (1.f / 256.f);
    float vs = 0.f;
    for (int c = l; c < 256; c += 32) { float d = sr[c] - m; vs += d * d; }
    for (int o = 1; o < 32; o <<= 1) vs += __shfl_xor(vs, o, 32);
    float inv = 1.f / sqrtf(vs * (1.f / 256.f) + 1e-5f);
    for (int c = l; c < 256; c += 32) {
      float x = (sr[c] - m) * inv * g[c] + bb[c];
      slnw[row * 256 + c] = x;
      if (c < cD) smuc[row * cD + c] = x;     // s_mu (GRU hidden state)
    }
  }
  for (int idx = t; idx < 64 * cD; idx += 256) { T1[idx] = 0.f; T2[idx] = 0.f; }
  for (int idx = t; idx < cB * 16; idx += 256) rsum[idx] = 0.f;
}

// ------------------------------------------------ q projections (WMMA)
__global__ void k_qproj(const float* __restrict__ slnw, const float* __restrict__ Wq,
                        float* __restrict__ qmuw, float* __restrict__ qlsw) {
  const int l = threadIdx.x & 31, wv = threadIdx.x >> 5;
  const int hl = l >> 4, sub = l & 15;
  int tid = blockIdx.x * 8 + wv;            // 0..63
  int which = tid & 1, nt = (tid >> 1) & 7, mt = tid >> 4;
  v8f c = gemm_xWt(slnw + mt * 16 * 256 + which * cD, 256,
                   Wq + nt * 16 * cD, cD, cD);
  float* dst = (which ? qlsw : qmuw) + mt * 16 * cD + nt * 16;
  for (int r = 0; r < 8; r++)
    dst[(r + hl * 8) * cD + sub] = c[r];
}

// ------------------------- build B' matrix for the dots GEMM + cterm
__global__ void k_dotsB(const float* __restrict__ qmuw, const float* __restrict__ qlsw,
                        float* __restrict__ dotsB, float* __restrict__ cterm) {
  const int t = threadIdx.x, wv = t >> 5, l = t & 31;
  for (int row = wv; row < 64; row += 8) {
    int b = row >> 3, i = row & 7;
    float acc = 0.f;
    for (int d = l; d < cD; d += 32) {
      float qm = qmuw[row * cD + d];
      float iv = expf(-2.f * qlsw[row * cD + d]);
      dotsB[(b * 256 + d) * 16 + i]       = cSCALE * iv;          // k^2 coeff
      dotsB[(b * 256 + cD + d) * 16 + i]  = -2.f * cSCALE * qm * iv; // k coeff
      acc += qm * qm * iv;
    }
    for (int o = 1; o < 32; o <<= 1) acc += __shfl_xor(acc, o, 32);
    if (l == 0) cterm[b * 16 + i] = cSCALE * acc;
  }
  for (int idx = t; idx < cB * 256 * 16; idx += 256)
    if ((idx & 15) >= cNS) dotsB[idx] = 0.f;   // pad slot columns
  for (int idx = t; idx < cB * 16; idx += 256)
    if ((idx & 15) >= cNS) cterm[idx] = 1e30f; // exp(-inf) -> 0 on pads
}

// --------------------- dots GEMM (A' = [k^2 ; k]) + attention normalization
__global__ void k_dots(const float* __restrict__ kbuf, const float* __restrict__ dotsB,
                       const float* __restrict__ cterm, const float* __restrict__ piw,
                       float* __restrict__ attnU, float* __restrict__ rsum) {
  __shared__ float tl[8][16][17];
  __shared__ float rs[8][16];
  __shared__ float colsum[16];
  const int t = threadIdx.x;
  const int wv = t >> 5, l = t & 31, hl = l >> 4, sub = l & 15;
  const int b = blockIdx.x >> 5;
  const int jb = (blockIdx.x & 31) * 128 + wv * 16;
  const float* kr = kbuf + ((size_t)b * cN + jb + sub) * cD + 2 * hl;
  const float* Bm = dotsB + b * 256 * 16 + sub;
  v8f c = zero8();
  for (int kk = 0; kk < cD; kk += 4) {          // k^2 half of K'
    v2f a = *(const v2f*)(kr + kk);
    a.x *= a.x; a.y *= a.y;
    int K0 = (kk + 2 * hl) * 16;
    v2f bb; bb.x = Bm[K0]; bb.y = Bm[K0 + 16];
    c = wmma4(a, bb, c);
  }
  for (int kk = 0; kk < cD; kk += 4) {          // raw k half of K'
    v2f a = *(const v2f*)(kr + kk);
    int K0 = (cD + kk + 2 * hl) * 16;
    v2f bb; bb.x = Bm[K0]; bb.y = Bm[K0 + 16];
    c = wmma4(a, bb, c);
  }
  float ct = cterm[b * 16 + sub];
  float pv = piw[b * 16 + sub];
  for (int r = 0; r < 8; r++) {
    int m = r + hl * 8;
    tl[wv][m][sub] = (expf(-(c[r] + ct)) + cEPS) * pv;
  }
  if (t < 16) colsum[t] = 0.f;
  __syncthreads();
  for (int idx = t; idx < 128; idx += 256) {    // per-row (per-j) slot sums
    int w2 = idx >> 4, rr = idx & 15;
    float s = 0.f;
#pragma unroll
    for (int q = 0; q < 16; q++) s += tl[w2][rr][q];
    rs[w2][rr] = s;
  }
  __syncthreads();
  const size_t jcol = (size_t)(blockIdx.x & 31) * 128;
  for (int idx = t; idx < 2048; idx += 256) {
    int w2 = idx >> 8, rr = (idx >> 4) & 15, nn = idx & 15;
    float a = tl[w2][rr][nn] / rs[w2][rr];
    attnU[((size_t)b * 16 + nn) * cN + jcol + w2 * 16 + rr] = a;
    atomicAdd(&colsum[nn], a);
  }
  __syncthreads();
  if (t < 16) atomicAdd(&rsum[b * 16 + t], colsum[t]);
}

// -------- moments: T1 = attnU @ v, T2 = attnU @ v^2 (single pass over v)
__global__ void k_moments(const float* __restrict__ attnU, const float* __restrict__ vbuf,
                          float* __restrict__ T1, float* __restrict__ T2) {
  const int t = threadIdx.x;
  const int wv = t >> 5, l = t & 31, hl = l >> 4, sub = l & 15;
  const int b = blockIdx.x >> 3;
  const int chunk = blockIdx.x & 7;             // K-split over j
  const int colb = wv * 16;
  const float* ar = attnU + ((size_t)b * 16 + sub) * cN + chunk * 512 + 2 * hl;
  const float* vb = vbuf + ((size_t)b * cN + chunk * 512 + 2 * hl) * cD + colb + sub;
  v8f c1 = zero8(), c2 = zero8();
  for (int kk = 0; kk < 512; kk += 4) {
    v2f a = *(const v2f*)(ar + kk);
    float v0 = vb[(size_t)kk * cD];
    float v1 = vb[(size_t)(kk + 1) * cD];
    v2f bv; bv.x = v0;      bv.y = v1;
    v2f bw; bw.x = v0 * v0; bw.y = v1 * v1;
    c1 = wmma4(a, bv, c1);
    c2 = wmma4(a, bw, c2);
  }
  if (hl == 0) {                                 // rows 0..7 = real slots
    for (int r = 0; r < 8; r++) {
      atomicAdd(&T1[((size_t)b * 8 + r) * cD + colb + sub], c1[r]);
      atomicAdd(&T2[((size_t)b * 8 + r) * cD + colb + sub], c2[r]);
    }
  }
}

// ------------------------------------------------ S1 = T1 / Rsum
__global__ void k_upd0(const float* __restrict__ T1, const float* __restrict__ rsum,
                       float* __restrict__ upd0w) {
  int t = blockIdx.x * 256 + threadIdx.x;
  if (t < 64 * cD) {
    int row = t >> 7;
    int b = row >> 3, i = row & 7;
    upd0w[t] = T1[t] / rsum[b * 16 + i];
  }
}

// ------------------------------------------------ GRU r,z gates
__global__ void k_gru_rz(const float* __restrict__ upd0w, const float* __restrict__ smuc,
                         const float* __restrict__ wih, const float* __restrict__ whh,
                         const float* __restrict__ bih, const float* __restrict__ bhh,
                         float* __restrict__ rgw, float* __restrict__ zgw) {
  const int l = threadIdx.x & 31, wv = threadIdx.x >> 5;
  const int hl = l >> 4, sub = l & 15;
  int tid = blockIdx.x * 8 + wv;                // 0..63
  int gate = tid >> 5, mt = (tid >> 3) & 3, nt = tid & 7;
  int od = gate * cD + nt * 16;
  v8f gi = gemm_xWt(upd0w + mt * 16 * cD, cD, wih + od * cD, cD, cD);
  v8f gh = gemm_xWt(smuc + mt * 16 * cD, cD, whh + od * cD, cD, cD);
  float bsum = bih[od + sub] + bhh[od + sub];
  float* dst = (gate ? zgw : rgw) + mt * 16 * cD + nt * 16;
  for (int r = 0; r < 8; r++) {
    float x = gi[r] + gh[r] + bsum;
    dst[(r + hl * 8) * cD + sub] = 1.f / (1.f + expf(-x));
  }
}

// ------------------------------------------------ GRU n gate + new hidden
__global__ void k_gru_n(const float* __restrict__ upd0w, const float* __restrict__ smuc,
                        const float* __restrict__ wih, const float* __restrict__ whh,
                        const float* __restrict__ bih, const float* __restrict__ bhh,
                        const float* __restrict__ rgw, const float* __restrict__ zgw,
                        float* __restrict__ mu1w) {
  const int l = threadIdx.x & 31, wv = threadIdx.x >> 5;
  const int hl = l >> 4, sub = l & 15;
  int tid = blockIdx.x * 8 + wv;                // 0..31
  int mt = tid >> 3, nt = tid & 7;
  int od = 2 * cD + nt * 16;
  v8f gi = gemm_xWt(upd0w + mt * 16 * cD, cD, wih + od * cD, cD, cD);
  v8f gh = gemm_xWt(smuc + mt * 16 * cD, cD, whh + od * cD, cD, cD);
  float bi = bih[od + sub], bh = bhh[od + sub];
  for (int r = 0; r < 8; r++) {
    int row = mt * 16 + r + hl * 8;
    int dd = nt * 16 + sub;
    float rr = rgw[row * cD + dd];
    float zz = zgw[row * cD + dd];
    float nn = tanhf(gi[r] + bi + rr * (gh[r] + bh));
    mu1w[row * cD + dd] = (1.f - zz) * nn + zz * smuc[row * cD + dd];
  }
}

// ------------------------------------------------ LN(mu1) for the MLP
__global__ void k_ln_mu(const float* __restrict__ mu1w, const float* __restrict__ g,
                        const float* __restrict__ bb, float* __restrict__ lnmw) {
  const int t = threadIdx.x, wv = t >> 5, l = t & 31;
  for (int row = wv; row < 64; row += 8) {
    const float* sr = mu1w + row * cD;
    float s = 0.f;
    for (int c = l; c < cD; c += 32) s += sr[c];
    for (int o = 1; o < 32; o <<= 1) s += __shfl_xor(s, o, 32);
    float m = s * (1.f / cD);
    float vs = 0.f;
    for (int c = l; c < cD; c += 32) { float d = sr[c] - m; vs += d * d; }
    for (int o = 1; o < 32; o <<= 1) vs += __shfl_xor(vs, o, 32);
    float inv = 1.f / sqrtf(vs * (1.f / cD) + 1e-5f);
    for (int c = l; c < cD; c += 32)
      lnmw[row * cD + c] = (sr[c] - m) * inv * g[c] + bb[c];
  }
}

// ------------------------------------------------ MLP hidden
__global__ void k_mlp1(const float* __restrict__ lnmw, const float* __restrict__ w1,
                       const float* __restrict__ b1, float* __restrict__ h1w) {
  const int l = threadIdx.x & 31, wv = threadIdx.x >> 5;
  const int hl = l >> 4, sub = l & 15;
  int tid = blockIdx.x * 8 + wv;                // 0..31
  int mt = tid >> 3, nt = tid & 7;
  v8f c = gemm_xWt(lnmw + mt * 16 * cD, cD, w1 + nt * 16 * cD, cD, cD);
  float bb = b1[nt * 16 + sub];
  for (int r = 0; r < 8; r++)
    h1w[(mt * 16 + r + hl * 8) * cD + nt * 16 + sub] = fmaxf(c[r] + bb, 0.f);
}

// ---------------- MLP out + residual -> final mu, logsigma, slots, pi
__global__ void k_mlp2(const float* __restrict__ h1w, const float* __restrict__ w2,
                       const float* __restrict__ b2, const float* __restrict__ mu1w,
                       const float* __restrict__ upd0w, const float* __restrict__ T2,
                       const float* __restrict__ rsum, float* __restrict__ slots,
                       float* __restrict__ piw) {
  const int l = threadIdx.x & 31, wv = threadIdx.x >> 5;
  const int hl = l >> 4, sub = l & 15;
  int tid = blockIdx.x * 8 + wv;                // 0..31
  int mt = tid >> 3, nt = tid & 7;
  v8f c = gemm_xWt(h1w + mt * 16 * cD, cD, w2 + nt * 16 * cD, cD, cD);
  float bb = b2[nt * 16 + sub];
  for (int r = 0; r < 8; r++) {
    int row = mt * 16 + r + hl * 8;
    int dd = nt * 16 + sub;
    int b = row >> 3, i = row & 7;
    float mu = mu1w[row * cD + dd] + c[r] + bb;
    float S1 = upd0w[row * cD + dd];
    float S2 = T2[row * cD + dd] / rsum[b * 16 + i];
    float val = S2 - 2.f * mu * S1 + mu * mu + cEPS;  // E[(v-mu)^2] + EPS
    slots[row * 256 + dd] = mu;
    slots[row * 256 + cD + dd] = 0.5f * logf(fmaxf(val, 1e-30f));
  }
  if (blockIdx.x == 0 && threadIdx.x < cB * 16)
    piw[threadIdx.x] = ((threadIdx.x & 15) < cNS) ? (1.0f / (float)cNS) : 0.f;
}

// ------------------------------------------------ final output MLP
__global__ void k_out1(const float* __restrict__ slots, const float* __restrict__ w1,
                       const float* __restrict__ b1, float* __restrict__ h1ow) {
  const int l = threadIdx.x & 31, wv = threadIdx.x >> 5;
  const int hl = l >> 4, sub = l & 15;
  int tid = blockIdx.x * 8 + wv;                // 0..63
  int mt = tid >> 4, nt = tid & 15;
  v8f c = gemm_xWt(slots + mt * 16 * 256, 256, w1 + nt * 16 * 256, 256, 256);
  float bb = b1[nt * 16 + sub];
  for (int r = 0; r < 8; r++)
    h1ow[(mt * 16 + r + hl * 8) * 256 + nt * 16 + sub] = fmaxf(c[r] + bb, 0.f);
}

__global__ void k_out2(const float* __restrict__ h1ow, const float* __restrict__ w2,
                       const float* __restrict__ b2, float* __restrict__ out) {
  const int l = threadIdx.x & 31, wv = threadIdx.x >> 5;
  const int hl = l >> 4, sub = l & 15;
  int tid = blockIdx.x * 8 + wv;                // 0..31
  int mt = tid >> 3, nt = tid & 7;
  v8f c = gemm_xWt(h1ow + mt * 16 * 256, 256, w2 + nt * 16 * 256, 256, 256);
  float bb = b2[nt * 16 + sub];
  for (int r = 0; r < 8; r++)
    out[(mt * 16 + r + hl * 8) * cD + nt * 16 + sub] = c[r] + bb;
}

extern "C" void kernel_launch(void* const* d_in, const int* in_sizes, int n_in,
                              void* d_out, int out_size, void* d_ws, size_t ws_size,
                              hipStream_t stream) {
  (void)in_sizes; (void)n_in; (void)out_size; (void)ws_size;
  const float* inputs   = (const float*)d_in[0];
  const float* noise    = (const float*)d_in[1];
  const float* slots_mu = (const float*)d_in[2];
  const float* slots_ls = (const float*)d_in[3];
  const float* Wq       = (const float*)d_in[4];
  const float* Wk       = (const float*)d_in[5];
  const float* Wv       = (const float*)d_in[6];
  const float* gru_wih  = (const float*)d_in[7];
  const float* gru_whh  = (const float*)d_in[8];
  const float* gru_bih  = (const float*)d_in[9];
  const float* gru_bhh  = (const float*)d_in[10];
  const float* mlp_w1   = (const float*)d_in[11];
  const float* mlp_b1   = (const float*)d_in[12];
  const float* mlp_w2   = (const float*)d_in[13];
  const float* mlp_b2   = (const float*)d_in[14];
  const float* out_w1   = (const float*)d_in[15];
  const float* out_b1   = (const float*)d_in[16];
  const float* out_w2   = (const float*)d_in[17];
  const float* out_b2   = (const float*)d_in[18];
  const float* ln_in_g  = (const float*)d_in[19];
  const float* ln_in_b  = (const float*)d_in[20];
  const float* ln_s_g   = (const float*)d_in[21];
  const float* ln_s_b   = (const float*)d_in[22];
  const float* ln_mu_g  = (const float*)d_in[23];
  const float* ln_mu_b  = (const float*)d_in[24];

  float* ws    = (float*)d_ws;
  float* kbuf  = ws;                     // 8*4096*128
  float* vbuf  = kbuf  + 4194304;        // 8*4096*128
  float* slots = vbuf  + 4194304;        // 64*256
  float* slnw  = slots + 16384;          // 64*256
  float* smuc  = slnw  + 16384;          // 64*128
  float* qmuw  = smuc  + 8192;           // 64*128
  float* qlsw  = qmuw  + 8192;           // 64*128
  float* attnU = qlsw  + 8192;           // 8*16*4096
  float* dotsB = attnU + 524288;         // 8*256*16
  float* cterm = dotsB + 32768;          // 8*16
  float* piw   = cterm + 128;            // 8*16
  float* rsumw = piw   + 128;            // 8*16
  float* T1    = rsumw + 128;            // 64*128
  float* T2    = T1    + 8192;           // 64*128
  float* upd0w = T2    + 8192;           // 64*128
  float* rgw   = upd0w + 8192;           // 64*128
  float* zgw   = rgw   + 8192;           // 64*128
  float* mu1w  = zgw   + 8192;           // 64*128
  float* lnmw  = mu1w  + 8192;           // 64*128
  float* h1w   = lnmw  + 8192;           // 64*128
  float* h1ow  = h1w   + 8192;           // 64*256

  k_init<<<64, 256, 0, stream>>>(slots_mu, slots_ls, noise, slots, piw);
  k_ln_proj<<<cB * cN / 16, 256, 0, stream>>>(inputs, ln_in_g, ln_in_b,
                                              Wk, Wv, kbuf, vbuf);
  for (int step = 0; step < 4; step++) {   // ITERS=3 + stop-gradient step
    k_slot_ln<<<1, 256, 0, stream>>>(slots, ln_s_g, ln_s_b, slnw, smuc,
                                     T1, T2, rsumw);
    k_qproj<<<8, 256, 0, stream>>>(slnw, Wq, qmuw, qlsw);
    k_dotsB<<<1, 256, 0, stream>>>(qmuw, qlsw, dotsB, cterm);
    k_dots<<<cB * 32, 256, 0, stream>>>(kbuf, dotsB, cterm, piw, attnU, rsumw);
    k_moments<<<cB * 8, 256, 0, stream>>>(attnU, vbuf, T1, T2);
    k_upd0<<<32, 256, 0, stream>>>(T1, rsumw, upd0w);
    k_gru_rz<<<8, 256, 0, stream>>>(upd0w, smuc, gru_wih, gru_whh,
                                    gru_bih, gru_bhh, rgw, zgw);
    k_gru_n<<<4, 256, 0, stream>>>(upd0w, smuc, gru_wih, gru_whh,
                                   gru_bih, gru_bhh, rgw, zgw, mu1w);
    k_ln_mu<<<1, 256, 0, stream>>>(mu1w, ln_mu_g, ln_mu_b, lnmw);
    k_mlp1<<<4, 256, 0, stream>>>(lnmw, mlp_w1, mlp_b1, h1w);
    k_mlp2<<<4, 256, 0, stream>>>(h1w, mlp_w2, mlp_b2, mu1w, upd0w, T2,
                                  rsumw, slots, piw);
  }
  k_out1<<<8, 256, 0, stream>>>(slots, out_w1, out_b1, h1ow);
  k_out2<<<4, 256, 0, stream>>>(h1ow, out_w2, out_b2, (float*)d_out);
}